// CircularBasisLayer_86629490360986
// MI455X (gfx1250) — compile-verified
//
#include <hip/hip_runtime.h>

// CircularBasisLayer fused kernel for gfx1250 (MI455X).
//
// out[t, s*16 + r] = env(d) * (norm/D) * sin((r+1)*pi*d) * exp(coeff*(cos-off_s)^2)
// with d = D[id3[t]] / CUTOFF.
//
// Roofline: 2M x 128 x 4B = 1.024 GB of stores dominates (~45 us at 23.3 TB/s);
// inputs are ~18 MB and L2-resident. Batched rank-1 outer product => no shared
// K-reduction => WMMA inapplicable. Strategy:
//   * lane l of a wave produces out[t, 4l..4l+3]: one b128 store per lane
//     => every store instruction writes one full 512B-aligned output row
//   * NON-TEMPORAL stores (TH_STORE_NT) so the 1 GB write stream does not
//     evict the L2-resident D_ca gather table / id / cos streams
//   * wave-uniform scalar loads for id/cos/D (s_load path, KMcnt)
//   * global_prefetch_b8 keeps the input streams ahead of the wave

#define CUTOFF     6.0f
#define NORM_BES   0.09622504486493764f   /* sqrt(2 / 6^3) */
#define GAUSS_CO  -6.125f                 /* -0.5 / (2/7)^2 */
#define PI_F       3.14159265358979323846f

typedef float v4f __attribute__((ext_vector_type(4)));

__device__ __forceinline__ v4f triplet_row4(float D, float cp,
                                            float off, v4f fr) {
    float dsc  = D * (1.0f / CUTOFF);
    float dsc2 = dsc * dsc;
    float dsc4 = dsc2 * dsc2;
    float dsc5 = dsc4 * dsc;
    // env = 1 - 21 d^5 + 35 d^6 - 15 d^7   (p = 5 polynomial envelope)
    float env = fmaf(dsc5, fmaf(dsc, fmaf(-15.0f, dsc, 35.0f), -21.0f), 1.0f);
    env = (dsc < 1.0f) ? env : 0.0f;
    float pref = env * NORM_BES * __builtin_amdgcn_rcpf(D);
    float dd = cp - off;
    float g  = __expf(GAUSS_CO * dd * dd);
    float pg = pref * g;
    v4f v;
    v.x = pg * __sinf(fr.x * dsc);
    v.y = pg * __sinf(fr.y * dsc);
    v.z = pg * __sinf(fr.z * dsc);
    v.w = pg * __sinf(fr.w * dsc);
    return v;
}

__global__ __launch_bounds__(256)
void CircularBasisLayer_86629490360986_kernel(const float* __restrict__ D_ca,
                                              const float* __restrict__ cos_phi,
                                              const int*   __restrict__ id3,
                                              float*       __restrict__ out,
                                              int n_trip) {
    const int lane  = threadIdx.x & 31;
    const int wib   = __builtin_amdgcn_readfirstlane((int)(threadIdx.x >> 5));
    const int wpb   = (int)(blockDim.x >> 5);
    const int gwave = (int)blockIdx.x * wpb + wib;
    const int nwave = (int)gridDim.x * wpb;

    // lane -> (spherical s, radial group r0)
    const int   s   = lane >> 2;
    const int   r0  = (lane & 3) * 4;
    const float off = -1.0f + (2.0f / 7.0f) * (float)s;
    const v4f   fr  = { (float)(r0 + 1) * PI_F, (float)(r0 + 2) * PI_F,
                        (float)(r0 + 3) * PI_F, (float)(r0 + 4) * PI_F };

    v4f* __restrict__ outv = (v4f*)out;
    const int step = nwave * 8;              // 8 triplets per wave-iteration

    for (int c = gwave * 8; c + 7 < n_trip; c += step) {
        // gfx1250 global_prefetch_b8: keep the input streams ahead.
        int pf = c + 2 * step;
        if (pf + 7 < n_trip) {
            __builtin_prefetch(id3 + pf, 0, 0);
            __builtin_prefetch(cos_phi + pf, 0, 0);
        }

        int4   idA = *(const int4*)(id3 + c);        // wave-uniform 16B
        int4   idB = *(const int4*)(id3 + c + 4);
        float4 cpA = *(const float4*)(cos_phi + c);
        float4 cpB = *(const float4*)(cos_phi + c + 4);

        // Force SGPR indices -> 8 independent scalar gathers under one wait.
        int i0 = __builtin_amdgcn_readfirstlane(idA.x);
        int i1 = __builtin_amdgcn_readfirstlane(idA.y);
        int i2 = __builtin_amdgcn_readfirstlane(idA.z);
        int i3 = __builtin_amdgcn_readfirstlane(idA.w);
        int i4 = __builtin_amdgcn_readfirstlane(idB.x);
        int i5 = __builtin_amdgcn_readfirstlane(idB.y);
        int i6 = __builtin_amdgcn_readfirstlane(idB.z);
        int i7 = __builtin_amdgcn_readfirstlane(idB.w);
        float d0 = D_ca[i0], d1 = D_ca[i1], d2 = D_ca[i2], d3 = D_ca[i3];
        float d4 = D_ca[i4], d5 = D_ca[i5], d6 = D_ca[i6], d7 = D_ca[i7];

        v4f v0 = triplet_row4(d0, cpA.x, off, fr);
        v4f v1 = triplet_row4(d1, cpA.y, off, fr);
        v4f v2 = triplet_row4(d2, cpA.z, off, fr);
        v4f v3 = triplet_row4(d3, cpA.w, off, fr);
        v4f v4 = triplet_row4(d4, cpB.x, off, fr);
        v4f v5 = triplet_row4(d5, cpB.y, off, fr);
        v4f v6 = triplet_row4(d6, cpB.z, off, fr);
        v4f v7 = triplet_row4(d7, cpB.w, off, fr);

        long b = (long)c * 32 + lane;        // v4f index of this lane's chunk
        // Non-temporal: 1 GB write-once stream must not evict L2 inputs.
        __builtin_nontemporal_store(v0, outv + b);
        __builtin_nontemporal_store(v1, outv + b + 32);
        __builtin_nontemporal_store(v2, outv + b + 64);
        __builtin_nontemporal_store(v3, outv + b + 96);
        __builtin_nontemporal_store(v4, outv + b + 128);
        __builtin_nontemporal_store(v5, outv + b + 160);
        __builtin_nontemporal_store(v6, outv + b + 192);
        __builtin_nontemporal_store(v7, outv + b + 224);
    }

    // Tail (n_trip % 8 != 0): wave 0 handles remaining triplets one at a time.
    if (gwave == 0) {
        for (int t = n_trip & ~7; t < n_trip; ++t) {
            int   id = __builtin_amdgcn_readfirstlane(id3[t]);
            float cp = cos_phi[t];
            v4f v = triplet_row4(D_ca[id], cp, off, fr);
            __builtin_nontemporal_store(v, outv + (long)t * 32 + lane);
        }
    }
}

extern "C" void kernel_launch(void* const* d_in, const int* in_sizes, int n_in,
                              void* d_out, int out_size, void* d_ws, size_t ws_size,
                              hipStream_t stream) {
    (void)n_in; (void)out_size; (void)d_ws; (void)ws_size;
    const float* D_ca    = (const float*)d_in[0];
    const float* cos_phi = (const float*)d_in[1];
    const int*   id3     = (const int*)d_in[2];
    float*       out     = (float*)d_out;
    const int n_trip = in_sizes[1];   // N_TRIPLETS

    dim3 grid(1024), block(256);      // 8192 waves -> ~16 waves/SIMD32
    hipLaunchKernelGGL(CircularBasisLayer_86629490360986_kernel,
                       grid, block, 0, stream,
                       D_ca, cos_phi, id3, out, n_trip);
}